// Sparsemax_25348896981609
// MI455X (gfx1250) — compile-verified
//
#include <hip/hip_runtime.h>

// Sparsemax over rows of length 2048, fp32 (16x2048x2048 -> 32768 rows).
// One wave32 per row; row held in 64 VGPRs/lane; Newton threshold search
// (exact for the piecewise-linear sparsemax threshold) instead of a full sort.
// Row staged global -> LDS via gfx1250 async DMA (ASYNCcnt) in two 4KB phases
// reusing one LDS slice, keeping LDS at 16KB/block so occupancy is VGPR-bound.

#define ROWS_PER_BLOCK 4
#define THREADS        128           // 4 waves (wave32) per block
#define ROW_LEN        2048
#define CHUNKS         16            // float4 chunks per lane: 16 * 32 * 4 = 2048
#define PHASE_CHUNKS   8             // chunks per lane per staging phase (4KB/wave)

typedef float v4f __attribute__((ext_vector_type(4)));    // native vector for NT stores

__global__ __launch_bounds__(THREADS)
void sparsemax_rows_kernel(const float* __restrict__ X, float* __restrict__ Y, int nrows) {
    __shared__ float4 tile[ROWS_PER_BLOCK][ROW_LEN / 8];   // 4 * 256 * 16B = 16 KB

    const int wave = threadIdx.x >> 5;
    const int lane = threadIdx.x & 31;
    const int row  = blockIdx.x * ROWS_PER_BLOCK + wave;
    if (row >= nrows) return;                              // wave-uniform

    const float* __restrict__ rowp = X + (size_t)row * ROW_LEN;

    float4 x[CHUNKS];

    // ---- Stage row global -> LDS with async DMA, two phases over one slice ----
#pragma unroll
    for (int p = 0; p < 2; ++p) {
        if (p) {
            // Phase-1 register copies must finish before DMA overwrites the slice.
            asm volatile("s_wait_dscnt 0x0" ::: "memory");
        }
#pragma unroll
        for (int k = 0; k < PHASE_CHUNKS; ++k) {
            const int cc = k * 32 + lane;                  // chunk index within phase
            const int c  = p * 256 + cc;                   // chunk index within row
            const float* gp = rowp + 4 * c;
            // Low 32 bits of a generic pointer to __shared__ == LDS byte offset
            // (AMDGPU addrspacecast ABI: generic = aperture_hi32 | lds_offset).
            unsigned lp = (unsigned)(unsigned long long)(const void*)&tile[wave][cc];
            asm volatile("global_load_async_to_lds_b128 %0, %1, off"
                         :: "v"(lp), "v"(gp)
                         : "memory");
        }
        asm volatile("s_wait_asynccnt 0x0" ::: "memory");
        // LDS -> registers (wave-private slice; no barrier needed).
#pragma unroll
        for (int k = 0; k < PHASE_CHUNKS; ++k)
            x[p * PHASE_CHUNKS + k] = tile[wave][k * 32 + lane];
    }

    // ---- Row max (numerical-stability shift, matches reference) ----
    float m = -3.402823466e+38f;
#pragma unroll
    for (int k = 0; k < CHUNKS; ++k)
        m = fmaxf(m, fmaxf(fmaxf(x[k].x, x[k].y), fmaxf(x[k].z, x[k].w)));
#pragma unroll
    for (int s = 16; s >= 1; s >>= 1)
        m = fmaxf(m, __shfl_xor(m, s, 32));

#pragma unroll
    for (int k = 0; k < CHUNKS; ++k) {
        x[k].x -= m; x[k].y -= m; x[k].z -= m; x[k].w -= m;
    }

    // ---- Newton on f(t) = sum(relu(xs - t)) - 1, t in [-1, 0) ----
    // f convex, piecewise linear, decreasing; Newton from t0=-1 is monotone,
    // never overshoots, and lands exactly on tau once the active set stabilizes.
    // Candidates (xs > -1) number ~10 for N(0,1) rows -> 20 iters is 2x margin.
    float t = -1.0f;
#pragma unroll 1
    for (int it = 0; it < 20; ++it) {
        float g = 0.0f, cnt = 0.0f;                        // g = sum relu(xs-t), cnt = |active|
#pragma unroll
        for (int k = 0; k < CHUNKS; ++k) {
            float d0 = x[k].x - t, d1 = x[k].y - t, d2 = x[k].z - t, d3 = x[k].w - t;
            g   += fmaxf(d0, 0.0f) + fmaxf(d1, 0.0f) + fmaxf(d2, 0.0f) + fmaxf(d3, 0.0f);
            cnt += (float)(d0 > 0.0f) + (float)(d1 > 0.0f)
                 + (float)(d2 > 0.0f) + (float)(d3 > 0.0f);
        }
#pragma unroll
        for (int s = 16; s >= 1; s >>= 1) {
            g   += __shfl_xor(g,   s, 32);
            cnt += __shfl_xor(cnt, s, 32);
        }
        // Root of current linear piece: t' = t + (g - 1)/cnt.
        // cnt >= 1 always (max element xs=0 > t for t < 0, and t stays < 0).
        t = t + (g - 1.0f) / cnt;
    }

    // ---- Write output: max(xs - tau, 0); streaming (non-temporal) stores ----
    v4f* __restrict__ outp = (v4f*)(Y + (size_t)row * ROW_LEN);
#pragma unroll
    for (int k = 0; k < CHUNKS; ++k) {
        const int c = k * 32 + lane;
        float4 v = x[k];
        v4f o;
        o.x = fmaxf(v.x - t, 0.0f);
        o.y = fmaxf(v.y - t, 0.0f);
        o.z = fmaxf(v.z - t, 0.0f);
        o.w = fmaxf(v.w - t, 0.0f);
        __builtin_nontemporal_store(o, outp + c);
    }
}

extern "C" void kernel_launch(void* const* d_in, const int* in_sizes, int n_in,
                              void* d_out, int out_size, void* d_ws, size_t ws_size,
                              hipStream_t stream) {
    const float* X = (const float*)d_in[0];
    float* Y = (float*)d_out;
    const int nrows  = in_sizes[0] / ROW_LEN;              // 16*2048 = 32768
    const int blocks = (nrows + ROWS_PER_BLOCK - 1) / ROWS_PER_BLOCK;
    sparsemax_rows_kernel<<<blocks, THREADS, 0, stream>>>(X, Y, nrows);
}